// KernelAttentionModel_65481071396579
// MI455X (gfx1250) — compile-verified
//
#include <hip/hip_runtime.h>

typedef __attribute__((ext_vector_type(2))) float v2f;
typedef __attribute__((ext_vector_type(4))) float v4f;
typedef __attribute__((ext_vector_type(8))) float v8f;

#define NN 1024
#define MM 128
#define DD 128

// ---------------------------------------------------------------------------
// Kernel 0: row norms  ra[j] = sum_k a[j][k]^2,  rb[n] = sum_k b[n][k]^2
// ---------------------------------------------------------------------------
__global__ void qk_norms_kernel(const float* __restrict__ a,
                                const float* __restrict__ b,
                                float* __restrict__ ra,
                                float* __restrict__ rb) {
    int t = blockIdx.x * blockDim.x + threadIdx.x;
    if (t < NN) {
        const v4f* row = (const v4f*)(a + (size_t)t * DD);
        float s = 0.f;
        #pragma unroll 8
        for (int i = 0; i < DD / 4; ++i) {
            v4f v = row[i];
            s += v.x * v.x + v.y * v.y + v.z * v.z + v.w * v.w;
        }
        ra[t] = s;
    } else if (t < NN + MM) {
        int r = t - NN;
        const v4f* row = (const v4f*)(b + (size_t)r * DD);
        float s = 0.f;
        #pragma unroll 8
        for (int i = 0; i < DD / 4; ++i) {
            v4f v = row[i];
            s += v.x * v.x + v.y * v.y + v.z * v.z + v.w * v.w;
        }
        rb[r] = s;
    }
}

// ---------------------------------------------------------------------------
// Kernel 1: K[j][n] = exp(2*(a_j . b_n) - ra[j] - rb[n])  via WMMA f32 16x16x4
// One wave per 16x16 output tile; 64 row-tiles x 8 col-tiles = 512 waves.
// ---------------------------------------------------------------------------
__global__ void qk_gram_kernel(const float* __restrict__ a,
                               const float* __restrict__ b,
                               const float* __restrict__ ra,
                               const float* __restrict__ rb,
                               float* __restrict__ Kmat) {
    int wave = (blockIdx.x * blockDim.x + threadIdx.x) >> 5;   // 0..511
    int lane = threadIdx.x & 31;
    int tm = wave >> 3;          // 0..63  (row tile)
    int tn = wave & 7;           // 0..7   (col tile)
    int half = lane >> 4;        // 0/1
    int lm   = lane & 15;

    // A 16x4 f32: lane half*16+lm supplies A[lm][k0 + 2*half + {0,1}]
    const float* arow = a + (size_t)(tm * 16 + lm) * DD + 2 * half;
    // B 4x16 f32: lane half*16+n supplies B[k0 + 2*half + {0,1}][n] = b[n][k...]
    const float* brow = b + (size_t)(tn * 16 + lm) * DD + 2 * half;

    v8f c = {};
    for (int k0 = 0; k0 < DD; k0 += 4) {
        v2f av = *(const v2f*)(arow + k0);
        v2f bv = *(const v2f*)(brow + k0);
        // 8 args: (neg_a, A, neg_b, B, c_mod, C, reuse_a, reuse_b)
        c = __builtin_amdgcn_wmma_f32_16x16x4_f32(
                false, av, false, bv, (short)0, c, false, false);
    }

    // C/D layout: VGPR r -> M = r + 8*half, N = lm
    int n = tn * 16 + lm;
    float rbn = rb[n];
    #pragma unroll
    for (int r = 0; r < 8; ++r) {
        int m = tm * 16 + half * 8 + r;
        // GAMMA = 1:  exp(-(ra+rb-2*dot)) = exp(2*dot - ra - rb)
        Kmat[(size_t)m * MM + n] = expf(2.0f * c[r] - ra[m] - rbn);
    }
}

// ---------------------------------------------------------------------------
// Kernel 2: out[i][j][k] = Kmat[j][k] * a[i][k]   (512 MiB streaming write)
// grid = (1024, 4); block = 256. Each thread's k-quad of a[i] is loop
// invariant (one v4f register); K rows stream from L2; stores are
// nontemporal b128 so the output stream doesn't evict K/a from cache.
// ---------------------------------------------------------------------------
__global__ void qk_bcast_kernel(const float* __restrict__ a,
                                const float* __restrict__ Kmat,
                                float* __restrict__ out) {
    int i  = blockIdx.x;           // 0..1023
    int jc = blockIdx.y;           // 0..3  (256 j values each)
    int t  = threadIdx.x;          // 0..255
    int k4 = t & 31;               // fixed v4f column per thread

    v4f av = ((const v4f*)(a + (size_t)i * DD))[k4];
    const v4f* K4   = (const v4f*)Kmat;
    v4f*       out4 = (v4f*)out + (size_t)i * (NN * DD / 4);

    int jbase = jc * 256 + (t >> 5);
    #pragma unroll 4
    for (int it = 0; it < 32; ++it) {
        int j = jbase + it * 8;                       // absolute j
        v4f kv = K4[(size_t)j * (DD / 4) + k4];
        v4f r = kv * av;                              // elementwise
        __builtin_nontemporal_store(r, out4 + (size_t)j * (DD / 4) + k4);
    }
}

// ---------------------------------------------------------------------------
extern "C" void kernel_launch(void* const* d_in, const int* in_sizes, int n_in,
                              void* d_out, int out_size, void* d_ws, size_t ws_size,
                              hipStream_t stream) {
    const float* a = (const float*)d_in[0];   // [1024,128]
    const float* b = (const float*)d_in[1];   // [128,128]
    // d_in[2] (rotation_params) and d_in[3] (entangle_params) are dead:
    // softmax over a length-1 axis is identically 1, so attn_weights == a.

    float* Kmat = (float*)d_ws;               // 1024*128 floats
    float* ra   = Kmat + NN * MM;             // 1024 floats
    float* rb   = ra + NN;                    // 128 floats
    float* out  = (float*)d_out;              // [1024,1024,128]

    qk_norms_kernel<<<dim3((NN + MM + 255) / 256), dim3(256), 0, stream>>>(a, b, ra, rb);
    qk_gram_kernel<<<dim3(64), dim3(256), 0, stream>>>(a, b, ra, rb, Kmat);
    qk_bcast_kernel<<<dim3(1024, 4), dim3(256), 0, stream>>>(a, Kmat, out);
}